// GATNET_2147483648575
// MI455X (gfx1250) — compile-verified
//
#include <hip/hip_runtime.h>
#include <hip/hip_bf16.h>

// ---------------------------------------------------------------------------
// GAT 2-layer forward for MI455X (gfx1250, wave32).
// GEMMs: V_WMMA_F32_16X16X4_F32, B panel staged in LDS (one tn strip/block).
// Edge phase: L2-resident gather/scatter with float atomics.
// ---------------------------------------------------------------------------

typedef float v2f __attribute__((ext_vector_type(2)));
typedef float v8f __attribute__((ext_vector_type(8)));

// ----------------------------- utility -------------------------------------

__global__ __launch_bounds__(256) void gat_fill_zero(float* __restrict__ p, long n) {
    long i = (long)blockIdx.x * blockDim.x + threadIdx.x;
    long stride = (long)gridDim.x * blockDim.x;
    for (; i < n; i += stride) p[i] = 0.0f;
}

// Ordered-key encoding so float max can use unsigned atomicMax.
__device__ __forceinline__ unsigned fkey(float f) {
    unsigned u = __float_as_uint(f);
    return (u & 0x80000000u) ? ~u : (u | 0x80000000u);
}
__device__ __forceinline__ float fdec(unsigned k) {
    return (k & 0x80000000u) ? __uint_as_float(k & 0x7fffffffu)
                             : __uint_as_float(~k);
}

// ------------------------ WMMA f32 GEMM (C = A*B) --------------------------
// A: [M,K] row-major, B: [K,NC] row-major, C: [M,NC] row-major. K, NC are
// compile-time so all addressing strength-reduces to immediate offsets.
// Block = 256 threads = 8 waves: one tn strip, 8 consecutive tm tiles.
// B panel [K x 16] (16 KB for K=256) is staged in LDS once per block.
//
// A 16x4 layout : lanes 0-15 -> M=lane,K={0,1}; lanes 16-31 -> M=lane-16,K={2,3}
// B 4x16 layout : lanes 0-15 -> N=lane,K={0,1}; lanes 16-31 -> N=lane-16,K={2,3}
// C 16x16 layout: vgpr v -> M=v (lanes 0-15) / M=v+8 (lanes 16-31), N=lane&15

template <int K, int NC>
__global__ __launch_bounds__(256) void gat_gemm_wmma_f32(
    const float* __restrict__ A, const float* __restrict__ B,
    float* __restrict__ C, int M) {
    constexpr int TN = NC / 16;
    __shared__ float Bs[K * 16];

    const int tn     = blockIdx.x % TN;
    const int tmBase = (blockIdx.x / TN) * 8;

    // cooperative stage of the B panel for this tn strip
    {
        const float* Bp = B + tn * 16;
        #pragma unroll
        for (int idx = 0; idx < K * 16; idx += 256) {
            int i = idx + threadIdx.x;
            Bs[i] = Bp[(i >> 4) * NC + (i & 15)];
        }
    }
    __syncthreads();

    const int tm = tmBase + (threadIdx.x >> 5);
    if (tm * 16 >= M) return;                       // wave-uniform: EXEC stays full

    const int lane = threadIdx.x & 31;
    const int half = lane >> 4;                     // 0: K pair {0,1}, 1: {2,3}
    const int r    = lane & 15;

    const float* arow = A  + (size_t)(tm * 16 + r) * K + 2 * half;
    const float* brow = Bs + (2 * half) * 16 + r;   // rows k+2h / k+2h+1, col r

    v8f c = {};
    #pragma unroll 4
    for (int k = 0; k < K; k += 4) {
        v2f a = *(const v2f*)(arow + k);            // A[M=r][k+2h .. k+2h+1]
        v2f b;
        b.x = brow[k * 16];                         // Bs[k+2h  ][r]
        b.y = brow[k * 16 + 16];                    // Bs[k+2h+1][r]
        c = __builtin_amdgcn_wmma_f32_16x16x4_f32(
            /*neg_a=*/false, a, /*neg_b=*/false, b,
            /*c_mod=*/(short)0, c, /*reuse_a=*/false, /*reuse_b=*/false);
    }

    float* crow = C + (size_t)(tm * 16 + 8 * half) * NC + tn * 16 + r;
    #pragma unroll
    for (int v = 0; v < 8; ++v) crow[v * NC] = c[v];
}

// ------------------- per-(node,head) attention reductions ------------------
// a_src[n,h] = sum_c h[n,h,c]*att_src[h,c] ; same for a_dst. C fixed at 64.

template <int H>
__global__ __launch_bounds__(256) void gat_att_reduce(
    const float* __restrict__ h, const float* __restrict__ att_s,
    const float* __restrict__ att_d, float* __restrict__ as_,
    float* __restrict__ ad_, int N) {
    int t = blockIdx.x * blockDim.x + threadIdx.x;
    if (t >= N * H) return;
    int n = t / H, hh = t % H;
    const float* hp = h + (size_t)n * H * 64 + hh * 64;
    const float* sp = att_s + hh * 64;
    const float* dp = att_d + hh * 64;
    float s = 0.0f, d = 0.0f;
    #pragma unroll
    for (int c = 0; c < 64; ++c) { float v = hp[c]; s += v * sp[c]; d += v * dp[c]; }
    as_[t] = s;
    ad_[t] = d;
}

__device__ __forceinline__ void edge_src_dst(const int* __restrict__ ei, int E,
                                             int e, int& s, int& d) {
    if (e < E) { s = ei[e]; d = ei[E + e]; }        // [2,E] row-major: src row, dst row
    else       { s = d = e - E; }                   // self loop
}

// ---------------- edge pass 1: segment max over destinations ---------------

template <int H>
__global__ __launch_bounds__(256) void gat_edge_max(
    const int* __restrict__ ei, int E, int N,
    const float* __restrict__ as_, const float* __restrict__ ad_,
    unsigned* __restrict__ mkey) {
    long t = (long)blockIdx.x * blockDim.x + threadIdx.x;
    if (t >= (long)(E + N) * H) return;
    int e = (int)(t / H), hh = (int)(t % H);
    int s, d; edge_src_dst(ei, E, e, s, d);
    float v = as_[s * H + hh] + ad_[d * H + hh];
    v = v > 0.0f ? v : 0.2f * v;                    // leaky_relu(0.2)
    atomicMax(&mkey[d * H + hh], fkey(v));
}

// --------------- edge pass 2: exp(e - m) and denominator sum ---------------

template <int H>
__global__ __launch_bounds__(256) void gat_edge_exp(
    const int* __restrict__ ei, int E, int N,
    const float* __restrict__ as_, const float* __restrict__ ad_,
    const unsigned* __restrict__ mkey, float* __restrict__ den,
    float* __restrict__ expb) {
    long t = (long)blockIdx.x * blockDim.x + threadIdx.x;
    if (t >= (long)(E + N) * H) return;
    int e = (int)(t / H), hh = (int)(t % H);
    int s, d; edge_src_dst(ei, E, e, s, d);
    float v = as_[s * H + hh] + ad_[d * H + hh];
    v = v > 0.0f ? v : 0.2f * v;
    float ex = expf(v - fdec(mkey[d * H + hh]));
    expb[t] = ex;                                   // t == e*H + hh
    atomicAdd(&den[d * H + hh], ex);
}

// ----------- edge pass 3: out[dst] += alpha * h[src] (scatter-add) ---------
// one thread per (edge, channel); head loop fully unrolled.

template <int H>
__global__ __launch_bounds__(256) void gat_edge_agg(
    const int* __restrict__ ei, int E, int N,
    const float* __restrict__ expb, const float* __restrict__ den,
    const float* __restrict__ hfeat, float* __restrict__ agg) {
    long t = (long)blockIdx.x * blockDim.x + threadIdx.x;
    if (t >= (long)(E + N) * 64) return;
    int e = (int)(t >> 6), c = (int)(t & 63);
    int s, d; edge_src_dst(ei, E, e, s, d);
    const float* hp = hfeat + (size_t)s * H * 64;
    float*       ap = agg   + (size_t)d * H * 64;
    #pragma unroll
    for (int hh = 0; hh < H; ++hh) {
        float w = expb[(size_t)e * H + hh] / (den[d * H + hh] + 1e-16f);
        atomicAdd(&ap[hh * 64 + c], w * hp[hh * 64 + c]);
    }
}

// ------------------------------- epilogues ---------------------------------

template <int F>    // F power of two
__global__ __launch_bounds__(256) void gat_elu_bias_inplace(
    float* __restrict__ x, const float* __restrict__ bias, long n) {
    long t = (long)blockIdx.x * blockDim.x + threadIdx.x;
    if (t >= n) return;
    float v = x[t] + bias[t & (F - 1)];
    x[t] = v > 0.0f ? v : (expf(v) - 1.0f);         // elu
}

template <int F>    // F power of two
__global__ __launch_bounds__(256) void gat_add_bias_out(
    const float* __restrict__ a, const float* __restrict__ bias,
    float* __restrict__ out, long n) {
    long t = (long)blockIdx.x * blockDim.x + threadIdx.x;
    if (t >= n) return;
    out[t] = a[t] + bias[t & (F - 1)];
}

// ------------------------------- host side ---------------------------------

extern "C" void kernel_launch(void* const* d_in, const int* in_sizes, int n_in,
                              void* d_out, int out_size, void* d_ws, size_t ws_size,
                              hipStream_t stream) {
    const float* x        = (const float*)d_in[0];
    const int*   ei       = (const int*)  d_in[1];
    const float* W1       = (const float*)d_in[2];
    const float* att_src1 = (const float*)d_in[3];
    const float* att_dst1 = (const float*)d_in[4];
    const float* bias1    = (const float*)d_in[5];
    const float* W2       = (const float*)d_in[6];
    const float* att_src2 = (const float*)d_in[7];
    const float* att_dst2 = (const float*)d_in[8];
    const float* bias2    = (const float*)d_in[9];
    float*       out      = (float*)d_out;

    const int IN = 256, OUT = 64, H = 4;
    const int N    = in_sizes[0] / IN;
    const int E    = in_sizes[1] / 2;
    const int Etot = E + N;

    // ---- workspace carve-up (floats). Zero-init region first & contiguous.
    float* ws = (float*)d_ws;
    size_t off = 0;
    float*    agg1 = ws + off; off += (size_t)N * 256;   // -> becomes x1 in place
    float*    agg2 = ws + off; off += (size_t)N * 64;
    unsigned* m1   = (unsigned*)(ws + off); off += (size_t)N * H;
    float*    den1 = ws + off; off += (size_t)N * H;
    unsigned* m2   = (unsigned*)(ws + off); off += (size_t)N;
    float*    den2 = ws + off; off += (size_t)N;
    const long zeroCount = (long)off;                    // contiguous zero region
    float*    h1   = ws + off; off += (size_t)N * 256;
    float*    h2   = ws + off; off += (size_t)N * 64;
    float*    as1  = ws + off; off += (size_t)N * H;
    float*    ad1  = ws + off; off += (size_t)N * H;
    float*    as2  = ws + off; off += (size_t)N;
    float*    ad2  = ws + off; off += (size_t)N;
    float*    exp1 = ws + off; off += (size_t)Etot * H;
    float*    exp2 = ws + off; off += (size_t)Etot;

    const int B = 256;
    auto blocks = [](long n, int b) { return (unsigned)((n + b - 1) / b); };
    const int tilesM  = N >> 4;                          // 16-row tiles (N % 16 == 0)
    const int blocksM = (tilesM + 7) / 8;                // 8 tm tiles per block

    // 0) zero accumulators / max-keys / denoms (single contiguous region)
    gat_fill_zero<<<blocks(zeroCount, B), B, 0, stream>>>(ws, zeroCount);

    // ---------------- layer 1 (H=4, concat) ----------------
    gat_gemm_wmma_f32<256, 256><<<blocksM * (256 / 16), B, 0, stream>>>(x, W1, h1, N);
    gat_att_reduce<4><<<blocks((long)N * H, B), B, 0, stream>>>(h1, att_src1, att_dst1, as1, ad1, N);
    gat_edge_max<4><<<blocks((long)Etot * H, B), B, 0, stream>>>(ei, E, N, as1, ad1, m1);
    gat_edge_exp<4><<<blocks((long)Etot * H, B), B, 0, stream>>>(ei, E, N, as1, ad1, m1, den1, exp1);
    gat_edge_agg<4><<<blocks((long)Etot * 64, B), B, 0, stream>>>(ei, E, N, exp1, den1, h1, agg1);
    gat_elu_bias_inplace<256><<<blocks((long)N * 256, B), B, 0, stream>>>(agg1, bias1, (long)N * 256);

    // ---------------- layer 2 (H=1, mean over 1 head == identity) ----------
    gat_gemm_wmma_f32<256, 64><<<blocksM * (64 / 16), B, 0, stream>>>(agg1, W2, h2, N);
    gat_att_reduce<1><<<blocks((long)N, B), B, 0, stream>>>(h2, att_src2, att_dst2, as2, ad2, N);
    gat_edge_max<1><<<blocks((long)Etot, B), B, 0, stream>>>(ei, E, N, as2, ad2, m2);
    gat_edge_exp<1><<<blocks((long)Etot, B), B, 0, stream>>>(ei, E, N, as2, ad2, m2, den2, exp2);
    gat_edge_agg<1><<<blocks((long)Etot * 64, B), B, 0, stream>>>(ei, E, N, exp2, den2, h2, agg2);
    gat_add_bias_out<64><<<blocks((long)N * OUT, B), B, 0, stream>>>(agg2, bias2, out, (long)N * OUT);
}